// qmlp_2qbit_79182017069406
// MI455X (gfx1250) — compile-verified
//
#include <hip/hip_runtime.h>

typedef __attribute__((ext_vector_type(2))) float v2f;
typedef __attribute__((ext_vector_type(8))) float v8f;

#define NPAIR 28
#define NB_TOT 1024
#define KTOT (NPAIR * 16)      // 448
#define STRIDE 453             // 453 % 64 == 5, coprime with 64 -> conflict-free LDS rows

// pair tables: combinations(range(8),2) order
__device__ const int d_pa[NPAIR] = {0,0,0,0,0,0,0,1,1,1,1,1,1,2,2,2,2,2,3,3,3,3,4,4,4,5,5,6};
__device__ const int d_qa[NPAIR] = {1,2,3,4,5,6,7,2,3,4,5,6,7,3,4,5,6,7,4,5,6,7,5,6,7,6,7,7};

// ---------------- tiny complex-double helpers ----------------
struct cd { double r, i; };
__device__ inline cd cmul(cd a, cd b) { return cd{a.r*b.r - a.i*b.i, a.r*b.i + a.i*b.r}; }
__device__ inline cd cadd(cd a, cd b) { return cd{a.r + b.r, a.i + b.i}; }
__device__ inline cd cscale(cd a, double s) { return cd{a.r*s, a.i*s}; }

// ---------------- kernel 1: build 28 x (4x4) real weight matrices ----------------
// W_p[j][l] = 0.5 * Re( sum_i h_i * U[i][j] * conj(U[i][l]) ),  h = diag(1,-1,1,-1)
// U = CNOT(0,1) * CNOT(1,0) * (G (x) I),  G = Rz(t5)Ry(t4)Rx(t3)Rz(t2)Ry(t1)Rx(t0)
__global__ void build_w_kernel(const float* __restrict__ weight, float* __restrict__ Wg) {
    int p = threadIdx.x;
    if (p >= NPAIR) return;

    double t[6];
#pragma unroll
    for (int i = 0; i < 6; ++i) t[i] = (double)weight[p*6 + i];

    cd G[2][2] = {{{1.0,0.0},{0.0,0.0}},{{0.0,0.0},{1.0,0.0}}};
#pragma unroll
    for (int g = 0; g < 6; ++g) {
        double c = cos(0.5*t[g]), s = sin(0.5*t[g]);
        cd R[2][2];
        int kind = g % 3;            // 0:Rx 1:Ry 2:Rz
        if (kind == 0)      { R[0][0]=cd{c,0};  R[0][1]=cd{0,-s}; R[1][0]=cd{0,-s}; R[1][1]=cd{c,0}; }
        else if (kind == 1) { R[0][0]=cd{c,0};  R[0][1]=cd{-s,0}; R[1][0]=cd{s,0};  R[1][1]=cd{c,0}; }
        else                { R[0][0]=cd{c,-s}; R[0][1]=cd{0,0};  R[1][0]=cd{0,0};  R[1][1]=cd{c,s}; }
        cd T[2][2];
#pragma unroll
        for (int i2 = 0; i2 < 2; ++i2)
#pragma unroll
            for (int j2 = 0; j2 < 2; ++j2)
                T[i2][j2] = cadd(cmul(R[i2][0], G[0][j2]), cmul(R[i2][1], G[1][j2]));
#pragma unroll
        for (int i2 = 0; i2 < 2; ++i2)
#pragma unroll
            for (int j2 = 0; j2 < 2; ++j2) G[i2][j2] = T[i2][j2];
    }

    cd U1[4][4];
#pragma unroll
    for (int i = 0; i < 4; ++i)
#pragma unroll
        for (int j = 0; j < 4; ++j)
            U1[i][j] = ((i & 1) == (j & 1)) ? G[i >> 1][j >> 1] : cd{0.0, 0.0};

    const int CN10[4][4] = {{1,0,0,0},{0,0,0,1},{0,0,1,0},{0,1,0,0}};
    const int CN01[4][4] = {{1,0,0,0},{0,1,0,0},{0,0,0,1},{0,0,1,0}};
    cd Bm[4][4], U[4][4];
#pragma unroll
    for (int i = 0; i < 4; ++i)
#pragma unroll
        for (int j = 0; j < 4; ++j) {
            cd acc = {0.0, 0.0};
#pragma unroll
            for (int k = 0; k < 4; ++k) acc = cadd(acc, cscale(U1[k][j], (double)CN10[i][k]));
            Bm[i][j] = acc;
        }
#pragma unroll
    for (int i = 0; i < 4; ++i)
#pragma unroll
        for (int j = 0; j < 4; ++j) {
            cd acc = {0.0, 0.0};
#pragma unroll
            for (int k = 0; k < 4; ++k) acc = cadd(acc, cscale(Bm[k][j], (double)CN01[i][k]));
            U[i][j] = acc;
        }

    const double h[4] = {1.0, -1.0, 1.0, -1.0};
#pragma unroll
    for (int j = 0; j < 4; ++j)
#pragma unroll
        for (int l = 0; l < 4; ++l) {
            double w = 0.0;
#pragma unroll
            for (int i = 0; i < 4; ++i)
                w += h[i] * (U[i][j].r*U[i][l].r + U[i][j].i*U[i][l].i); // Re(U_ij conj(U_il))
            Wg[p*16 + j*4 + l] = (float)(0.5 * w);
        }
}

// ---------------- kernel 2: partial trace gather + WMMA contraction ----------------
// 64 blocks x 256 threads; each block handles 16 batch items.
__global__ __launch_bounds__(256) void qmlp_main_kernel(const float* __restrict__ x,
                                                        const float* __restrict__ Wg,
                                                        float* __restrict__ out) {
    __shared__ float Plds[16 * STRIDE];   // P[b_local][k = p*16 + j*4 + l]
    __shared__ float WLDS[KTOT];          // staged weight table

    const int btile = blockIdx.x;                  // batches [btile*16, btile*16+16)
    const int wave  = threadIdx.x >> 5;
    const int lane  = threadIdx.x & 31;

    // ---- async-stage the 448-float W table into LDS (ASYNCcnt path), overlapped
    // ---- with the entire stage-1 gather below.
    {
        const unsigned long long wbase = (unsigned long long)(uintptr_t)Wg;
        const int idx = threadIdx.x;                       // 0..255 < 448
        unsigned int lds0  = (unsigned int)(uintptr_t)(&WLDS[idx]);
        unsigned int goff0 = (unsigned int)idx * 4u;
        asm volatile("global_load_async_to_lds_b32 %0, %1, %2"
                     :: "v"(lds0), "v"(goff0), "s"(wbase) : "memory");
        if (idx < KTOT - 256) {                            // 256..447
            unsigned int lds1  = (unsigned int)(uintptr_t)(&WLDS[idx + 256]);
            unsigned int goff1 = (unsigned int)(idx + 256) * 4u;
            asm volatile("global_load_async_to_lds_b32 %0, %1, %2"
                         :: "v"(lds1), "v"(goff1), "s"(wbase) : "memory");
        }
    }

    // ---- stage 1: rho[b_local, p, j, l] = sum_{k=0..63} x[b, row(j,k), col(l,k)] ----
    // entries: 16 batches * 448 k-indices = 7168; one wave per entry, 2 gathers/lane + xor-reduce
    for (int e = wave; e < 16 * KTOT; e += 8) {
        const int bl  = e / KTOT;
        const int rem = e - bl * KTOT;
        const int p   = rem >> 4;
        const int m   = rem & 15;
        const int j   = m >> 2, l = m & 3;
        const int ba  = 7 - d_pa[p];               // bit position of qubit a (MSB-first)
        const int bq  = 7 - d_qa[p];               // bit position of qubit b; ba > bq
        const int rbits = ((j >> 1) << ba) | ((j & 1) << bq);
        const int cbits = ((l >> 1) << ba) | ((l & 1) << bq);
        const int base  = (rbits << 8) | cbits;
        const float* xb = x + ((size_t)(btile * 16 + bl) << 16) + base;

        const int mq = (1 << bq) - 1;
        const int ma = (1 << ba) - 1;
        float acc = 0.0f;
#pragma unroll
        for (int hh = 0; hh < 2; ++hh) {
            int k  = lane + 32 * hh;
            int t1 = ((k  & ~mq) << 1) | (k  & mq);  // insert 0 at bit bq
            int s  = ((t1 & ~ma) << 1) | (t1 & ma);  // insert 0 at bit ba
            acc += xb[s * 257];                      // row += s, col += s  (diagonal of traced qubits)
        }
        acc += __shfl_xor(acc, 16);
        acc += __shfl_xor(acc, 8);
        acc += __shfl_xor(acc, 4);
        acc += __shfl_xor(acc, 2);
        acc += __shfl_xor(acc, 1);
        if (lane == 0) Plds[bl * STRIDE + rem] = acc;
    }

    asm volatile("s_wait_asynccnt 0x0" ::: "memory");   // W table landed in LDS
    __syncthreads();

    // ---- stage 2: out[16 x 28] = P[16 x 448] @ Wbig[448 x 28] via v_wmma_f32_16x16x4_f32 ----
    // Wbig is block diagonal: column p only sees K in [p*16, p*16+16). Waves 0/1 own the two
    // 16-wide column tiles. Each lane owns column N = lane%16, i.e. pair p0+N; it preloads the
    // 8 W values it can ever contribute, so the WMMA loop is pure LDS + cndmask + wmma.
    if (wave < 2) {
        const int ntile = wave;
        const int p0    = ntile * 16;
        int pend        = p0 + 16; if (pend > NPAIR) pend = NPAIR;

        const int mrow = lane & 15;            // A: row M; B: column N (same lane mapping)
        const int koff = (lane >> 4) << 1;     // lanes 0-15 -> K0,K1 ; lanes 16-31 -> K2,K3

        const int  myp   = p0 + mrow;          // the pair whose output column this lane owns
        const bool valid = (myp < NPAIR);
        const int  wbase = (valid ? myp : 0) * 16;
        float wpre[8];
#pragma unroll
        for (int c = 0; c < 4; ++c) {
            wpre[c*2 + 0] = valid ? WLDS[wbase + c*4 + koff + 0] : 0.0f;
            wpre[c*2 + 1] = valid ? WLDS[wbase + c*4 + koff + 1] : 0.0f;
        }

        v8f acc = {0.f,0.f,0.f,0.f,0.f,0.f,0.f,0.f};
        for (int p = p0; p < pend; ++p) {
            const bool mycol = (mrow == (p - p0));
#pragma unroll
            for (int cch = 0; cch < 4; ++cch) {
                const int kbase = p * 16 + cch * 4;
                v2f a, b;
                a.x = Plds[mrow * STRIDE + kbase + koff];
                a.y = Plds[mrow * STRIDE + kbase + koff + 1];
                b.x = mycol ? wpre[cch*2 + 0] : 0.0f;
                b.y = mycol ? wpre[cch*2 + 1] : 0.0f;
                acc = __builtin_amdgcn_wmma_f32_16x16x4_f32(
                    /*neg_a=*/false, a, /*neg_b=*/false, b,
                    /*c_mod=*/(short)0, acc, /*reuse_a=*/false, /*reuse_b=*/false);
            }
        }

        // C/D layout: VGPR g -> M = g (lanes 0-15) or g+8 (lanes 16-31), N = lane%16
#pragma unroll
        for (int g = 0; g < 8; ++g) {
            const int M = g + ((lane >> 4) << 3);
            const int N = lane & 15;
            const int p = p0 + N;
            if (p < NPAIR)
                out[(size_t)(btile * 16 + M) * NPAIR + p] = 0.5f + acc[g];
        }
    }
}

extern "C" void kernel_launch(void* const* d_in, const int* in_sizes, int n_in,
                              void* d_out, int out_size, void* d_ws, size_t ws_size,
                              hipStream_t stream) {
    const float* x      = (const float*)d_in[0];   // [1024, 256, 256] f32
    const float* weight = (const float*)d_in[1];   // [168] f32
    float*       out    = (float*)d_out;           // [1024, 28] f32
    float*       Wg     = (float*)d_ws;            // 448 floats of scratch

    build_w_kernel<<<1, 32, 0, stream>>>(weight, Wg);
    qmlp_main_kernel<<<NB_TOT / 16, 256, 0, stream>>>(x, Wg, out);
}